// GCNLayer_9268539425407
// MI455X (gfx1250) — compile-verified
//
#include <hip/hip_runtime.h>

typedef __attribute__((ext_vector_type(2))) float v2f;
typedef __attribute__((ext_vector_type(8))) float v8f;

#define D 128            // D_IN == D_OUT == 128
#define NTILES_N (D/16)  // 8 N-tiles per 16-row output strip

// ---------------------------------------------------------------------------
// Zero the output buffer (atomics accumulate into it; harness poisons 0xAA).
// ---------------------------------------------------------------------------
__global__ void gcn_zero_kernel(float* __restrict__ out, int n4) {
    int i = blockIdx.x * blockDim.x + threadIdx.x;
    if (i < n4) {
        ((float4*)out)[i] = make_float4(0.f, 0.f, 0.f, 0.f);
    }
}

// ---------------------------------------------------------------------------
// support[M, 128] = embeds[M, 128] @ weight[128, 128]
//
// One wave32 per 16-row x 128-col output strip: A is read exactly once from
// HBM (global_load_b64 per lane per k-step). W is staged into LDS in a
// K-pair-interleaved layout: Wlds2[k/2][c] = {W[k][c], W[k+1][c]}, so each
// WMMA B fragment is a single ds_load_b64 into an adjacent (even-aligned)
// VGPR pair -- no register shuffles. 8 independent accumulator chains per
// wave; V_WMMA_F32_16X16X4_F32, K stepped by 4, fully unrolled (256 WMMA).
//
// Fragment layouts (ISA 7.12.2, wave32):
//   A 16x4  : VGPR0 = A[m][ka], VGPR1 = A[m][ka+1], ka = k + 2*(lane>>4)
//   B 4x16  : VGPR0 = W[ka][n], VGPR1 = W[ka+1][n],  n col = lane&15
//   C 16x16 : VGPR r = C[r + 8*(lane>>4)][lane&15]
// ---------------------------------------------------------------------------
__global__ void __launch_bounds__(256)
gcn_gemm_wmma_f32(const float* __restrict__ A,   // [M, D]
                  const float* __restrict__ W,   // [D, D]
                  float* __restrict__ S,         // [M, D]
                  int M) {
    __shared__ float2 Wlds2[(D / 2) * D];        // 64 KB, K-pair interleaved

    // Stage W with K-pair interleave: Wlds2[p*D + c] = {W[2p][c], W[2p+1][c]}.
    // Each iteration: two b128 row reads -> 32 contiguous bytes to LDS.
    for (int i = threadIdx.x; i < (D / 2) * (D / 4); i += blockDim.x) {
        const int p  = i / (D / 4);              // K row pair, 0..63
        const int c4 = (i % (D / 4)) * 4;        // column group
        const float4 r0 = *(const float4*)(W + (size_t)(2 * p)     * D + c4);
        const float4 r1 = *(const float4*)(W + (size_t)(2 * p + 1) * D + c4);
        float2* outp = Wlds2 + p * D + c4;
        outp[0] = make_float2(r0.x, r1.x);
        outp[1] = make_float2(r0.y, r1.y);
        outp[2] = make_float2(r0.z, r1.z);
        outp[3] = make_float2(r0.w, r1.w);
    }
    __syncthreads();

    const int lane  = threadIdx.x & 31;
    const int tileM = blockIdx.x * (blockDim.x >> 5) + (threadIdx.x >> 5);
    if (tileM * 16 >= M) return;                 // wave-uniform exit

    const int half = lane >> 4;                  // selects K pair {0,1} / {2,3}
    const int l15  = lane & 15;
    // Clamp (not predicate) the A row: loads always in-range, EXEC stays full.
    const int m    = min(tileM * 16 + l15, M - 1);
    const float*  Arow = A + (size_t)m * D;      // lane's A row base
    const float2* Wb   = Wlds2 + half * D + l15; // lane's B base (float2 units)

    v8f acc[NTILES_N] = {};
#pragma unroll
    for (int k = 0; k < D; k += 4) {
        const v2f a = *(const v2f*)(Arow + k + half * 2);    // global_load_b64
#pragma unroll
        for (int n = 0; n < NTILES_N; ++n) {
            // {W[ka][col], W[ka+1][col]} in one ds_load_b64, imm offset
            const v2f b = *(const v2f*)(Wb + (k >> 1) * D + n * 16);
            // (neg_a, A, neg_b, B, c_mod, C, reuse_a, reuse_b)
            acc[n] = __builtin_amdgcn_wmma_f32_16x16x4_f32(
                false, a, false, b, (short)0, acc[n], false, false);
        }
    }

    // Store the 16x128 strip. Divergence here is fine (all WMMAs retired).
#pragma unroll
    for (int r = 0; r < 8; ++r) {
        const int row = tileM * 16 + half * 8 + r;
        if (row < M) {
            float* Sp = S + (size_t)row * D + l15;
#pragma unroll
            for (int n = 0; n < NTILES_N; ++n) {
                Sp[n * 16] = acc[n][r];
            }
        }
    }
}

// ---------------------------------------------------------------------------
// out[dst[e]] += vals[e] * support[src[e]]   (segment-sum via f32 atomics)
// One wave32 per edge; each lane gathers float4 (global_load_b128) and
// issues 4 non-returning global_atomic_add_f32.
// ---------------------------------------------------------------------------
__global__ void gcn_spmm_scatter(const float* __restrict__ support, // [M, D]
                                 const int*   __restrict__ dst,     // [E]
                                 const int*   __restrict__ src,     // [E]
                                 const float* __restrict__ vals,    // [E]
                                 float*       __restrict__ out,     // [M, D]
                                 int E) {
    const int edge = (blockIdx.x * blockDim.x + threadIdx.x) >> 5;
    const int lane = threadIdx.x & 31;
    if (edge >= E) return;

    const int   s = src[edge];
    const int   d = dst[edge];
    const float v = vals[edge];

    const float4 msg = ((const float4*)(support + (size_t)s * D))[lane];
    float* op = out + (size_t)d * D + lane * 4;
    atomicAdd(op + 0, v * msg.x);
    atomicAdd(op + 1, v * msg.y);
    atomicAdd(op + 2, v * msg.z);
    atomicAdd(op + 3, v * msg.w);
}

// ---------------------------------------------------------------------------
extern "C" void kernel_launch(void* const* d_in, const int* in_sizes, int n_in,
                              void* d_out, int out_size, void* d_ws, size_t ws_size,
                              hipStream_t stream) {
    const float* embeds = (const float*)d_in[0];   // [M, 128]
    const float* weight = (const float*)d_in[1];   // [128, 128]
    const int*   eidx   = (const int*)d_in[2];     // [2, E]  row0=dst, row1=src
    const float* evals  = (const float*)d_in[3];   // [E]

    const int M = in_sizes[0] / D;
    const int E = in_sizes[3];
    const int* dst = eidx;
    const int* src = eidx + E;

    float* support = (float*)d_ws;                 // [M, 128] = 51.2 MB scratch
    float* out     = (float*)d_out;

    // 1) out = 0
    {
        const int n4 = out_size / 4;
        gcn_zero_kernel<<<(n4 + 255) / 256, 256, 0, stream>>>(out, n4);
    }

    // 2) support = embeds @ weight  (fp32 WMMA, one wave per 16-row strip)
    {
        const int mtiles = (M + 15) / 16;          // waves needed
        const int wavesPerBlock = 8;               // 256 threads
        const int blocks = (mtiles + wavesPerBlock - 1) / wavesPerBlock;
        gcn_gemm_wmma_f32<<<blocks, wavesPerBlock * 32, 0, stream>>>(
            embeds, weight, support, M);
    }

    // 3) scatter: out[dst] += vals * support[src]
    {
        const int wavesPerBlock = 8;               // 256 threads, 8 edges/block
        const int blocks = (E + wavesPerBlock - 1) / wavesPerBlock;
        gcn_spmm_scatter<<<blocks, wavesPerBlock * 32, 0, stream>>>(
            support, dst, src, evals, out, E);
    }
}